// ModelWrapper_43001212567606
// MI455X (gfx1250) — compile-verified
//
#include <hip/hip_runtime.h>
#include <hip/hip_bf16.h>
#include <stdint.h>

#define B_      16
#define N_      25200
#define NC_     80
#define ROW_    85
#define M_      32768       // padded per-batch sort length (2^15)
#define TOPK_   4096
#define MAXDET_ 1000
#define CONF_T  0.25f
#define IOU_T   0.45f
#define MAXWH_  7680.0f
#define NMS_THREADS 256
#define PER_T   (TOPK_ / NMS_THREADS)   // 16

typedef int v4i __attribute__((ext_vector_type(4)));
typedef __attribute__((address_space(1))) v4i v4i_g;   // global int4
typedef __attribute__((address_space(3))) v4i v4i_l;   // LDS int4

// ---------------------------------------------------------------------------
// Kernel 1: per-anchor confidence -> packed descending sort key.
// key = (monotonic_u32(score) << 32) | ~anchor_index  ; padding -> key 0.
// Equal scores tie-break to smaller index (matches JAX stable argsort).
// ---------------------------------------------------------------------------
__global__ void score_key_kernel(const float* __restrict__ pred,
                                 unsigned long long* __restrict__ keys) {
  int tid = blockIdx.x * blockDim.x + threadIdx.x;
  int b = tid >> 15;
  int n = tid & (M_ - 1);
  if (b >= B_) return;
  unsigned long long key = 0ull;
  if (n < N_) {
    const float* row = pred + ((long long)b * N_ + n) * ROW_;
    __builtin_prefetch(row, 0, 0);          // global_prefetch_b8 (gfx1250)
    float obj = row[4];
    float best = -1e30f;
    for (int j = 0; j < NC_; ++j) {
      float p = row[5 + j] * obj;           // scores_all = cls * obj (per-element rounding)
      best = fmaxf(best, p);
    }
    float s = (best > CONF_T) ? best : -1.0f;
    unsigned int u = __float_as_uint(s);
    unsigned int m = (u & 0x80000000u) ? ~u : (u | 0x80000000u);
    key = ((unsigned long long)m << 32) | (unsigned int)~n;
  }
  keys[(long long)b * M_ + n] = key;
}

// ---------------------------------------------------------------------------
// Kernel 2: one bitonic pass (descending sort), all batches per launch.
// 4 MB key array stays resident in the 192 MB L2 across the 120 passes.
// ---------------------------------------------------------------------------
__global__ void bitonic_pass_kernel(unsigned long long* __restrict__ keys,
                                    int k, int j) {
  int tid = blockIdx.x * blockDim.x + threadIdx.x;
  int b = tid >> 15;
  int i = tid & (M_ - 1);
  if (b >= B_) return;
  int ixj = i ^ j;
  if (ixj > i) {
    unsigned long long* base = keys + (long long)b * M_;
    unsigned long long a = base[i];
    unsigned long long c = base[ixj];
    bool up = ((i & k) == 0);
    if (up ? (a < c) : (a > c)) { base[i] = c; base[ixj] = a; }
  }
}

// ---------------------------------------------------------------------------
// Kernel 3: gather top-4096 per batch; recompute conf/argmax/boxes exactly.
// ---------------------------------------------------------------------------
__global__ void gather_topk_kernel(const float* __restrict__ pred,
                                   const unsigned long long* __restrict__ keys,
                                   float4* __restrict__ obA,
                                   float4* __restrict__ bxA,
                                   float*  __restrict__ scA,
                                   float*  __restrict__ clsA) {
  int tid = blockIdx.x * blockDim.x + threadIdx.x;
  int b = tid >> 12;
  int r = tid & (TOPK_ - 1);
  if (b >= B_) return;
  unsigned long long key = keys[(long long)b * M_ + r];
  float4 bx = make_float4(0.f, 0.f, 0.f, 0.f);
  float4 ob = make_float4(0.f, 0.f, 0.f, 0.f);
  float s = -1.0f, cf = 0.0f;
  if (key != 0ull) {
    unsigned int n = ~(unsigned int)key;
    const float* row = pred + ((long long)b * N_ + n) * ROW_;
    float obj = row[4];
    float best = -1e30f; int bi = 0;
    for (int j = 0; j < NC_; ++j) {
      float p = row[5 + j] * obj;
      if (p > best) { best = p; bi = j; }   // first-occurrence argmax (jnp.argmax)
    }
    s  = (best > CONF_T) ? best : -1.0f;
    cf = (float)bi;
    float x = row[0], y = row[1], w = row[2], h = row[3];
    bx = make_float4(x - w * 0.5f, y - h * 0.5f, x + w * 0.5f, y + h * 0.5f);
    float off = cf * MAXWH_;
    ob = make_float4(bx.x + off, bx.y + off, bx.z + off, bx.w + off);
  }
  obA[tid] = ob; bxA[tid] = bx; scA[tid] = s; clsA[tid] = cf;
}

// ---------------------------------------------------------------------------
// Kernel 4: greedy NMS, one workgroup per image. 4096 offset-boxes + scores
// live in LDS (CDNA5: 320 KB/WGP makes the whole candidate set resident).
// Staging uses gfx1250 async global->LDS loads tracked by ASYNCcnt.
// ---------------------------------------------------------------------------
__device__ __forceinline__ float iou4(float4 A, float4 Bx) {
  float lx = fmaxf(A.x, Bx.x), ly = fmaxf(A.y, Bx.y);
  float rx = fminf(A.z, Bx.z), ry = fminf(A.w, Bx.w);
  float w = fmaxf(rx - lx, 0.0f), h = fmaxf(ry - ly, 0.0f);
  float inter = w * h;
  float a1 = (A.z - A.x) * (A.w - A.y);
  float a2 = (Bx.z - Bx.x) * (Bx.w - Bx.y);
  return inter / (a1 + a2 - inter + 1e-7f);
}

__global__ __launch_bounds__(NMS_THREADS)
void nms_kernel(const float4* __restrict__ obA,
                const float4* __restrict__ bxA,
                const float*  __restrict__ scA,
                const float*  __restrict__ clsA,
                float* __restrict__ out) {
  extern __shared__ char smem[];
  float4* obS = (float4*)smem;                                       // 65536 B
  float*  scS = (float*)(smem + 65536);                              // 16384 B
  volatile unsigned char* supp =
      (volatile unsigned char*)(smem + 65536 + 16384);               //  4096 B
  unsigned char* keep = (unsigned char*)(smem + 65536 + 16384 + 4096); // 4096 B
  unsigned int* scanT = (unsigned int*)(smem + 65536 + 16384 + 8192);  // 1028 B

  const int b = blockIdx.x;
  const int t = threadIdx.x;
  const long long base = (long long)b * TOPK_;

  // zero this image's output block (harness poisons d_out)
  for (int q = t; q < MAXDET_ * 6; q += NMS_THREADS)
    out[(long long)b * MAXDET_ * 6 + q] = 0.0f;

#if defined(__gfx1250__) && __has_builtin(__builtin_amdgcn_global_load_async_to_lds_b128)
  for (int r = t; r < TOPK_; r += NMS_THREADS)
    __builtin_amdgcn_global_load_async_to_lds_b128(
        (v4i_g*)&obA[base + r], (v4i_l*)&obS[r], 0, 0);
  for (int r = t; r < TOPK_ / 4; r += NMS_THREADS)
    __builtin_amdgcn_global_load_async_to_lds_b128(
        (v4i_g*)&scA[base + r * 4], (v4i_l*)&scS[r * 4], 0, 0);
#if __has_builtin(__builtin_amdgcn_s_wait_asynccnt)
  __builtin_amdgcn_s_wait_asynccnt(0);
#else
  asm volatile("s_wait_asynccnt 0" ::: "memory");
#endif
#else
  for (int r = t; r < TOPK_; r += NMS_THREADS) obS[r] = obA[base + r];
  for (int r = t; r < TOPK_; r += NMS_THREADS) scS[r] = scA[base + r];
#endif
  for (int r = t; r < TOPK_; r += NMS_THREADS) { supp[r] = 0; keep[r] = 0; }
  __syncthreads();

  // greedy scan: candidate i alive <=> not suppressed and score > 0 (uniform
  // decision from LDS). Writes only set supp[j]=1 for j>i -> races benign.
  for (int i = 0; i < TOPK_; ++i) {
    bool alive = (!supp[i]) && (scS[i] > 0.0f);
    if (alive) {
      if (t == 0) keep[i] = 1;
      float4 bi = obS[i];
      int j0 = t * PER_T;
#pragma unroll
      for (int m2 = 0; m2 < PER_T; ++m2) {
        int j = j0 + m2;
        if (j > i && !supp[j]) {
          if (iou4(bi, obS[j]) > IOU_T) supp[j] = 1;
        }
      }
      __syncthreads();        // only alive iterations mutate supp
    }
  }
  __syncthreads();

  // compact kept rows in order (already score-descending) into out[b][0:1000]
  int j0 = t * PER_T, cnt = 0;
  for (int m2 = 0; m2 < PER_T; ++m2) cnt += keep[j0 + m2];
  scanT[t + 1] = (unsigned)cnt;
  if (t == 0) scanT[0] = 0u;
  __syncthreads();
  if (t == 0)
    for (int q = 1; q <= NMS_THREADS; ++q) scanT[q] += scanT[q - 1];
  __syncthreads();
  unsigned pos = scanT[t];
  for (int m2 = 0; m2 < PER_T; ++m2) {
    int i = j0 + m2;
    if (keep[i]) {
      if (pos < MAXDET_) {
        float4 bx = bxA[base + i];
        float* o = out + ((long long)b * MAXDET_ + pos) * 6;
        o[0] = bx.x; o[1] = bx.y; o[2] = bx.z; o[3] = bx.w;
        o[4] = scS[i]; o[5] = clsA[base + i];
      }
      ++pos;
    }
  }
}

// ---------------------------------------------------------------------------
extern "C" void kernel_launch(void* const* d_in, const int* in_sizes, int n_in,
                              void* d_out, int out_size, void* d_ws, size_t ws_size,
                              hipStream_t stream) {
  const float* pred = (const float*)d_in[0];
  float* out = (float*)d_out;
  char* ws = (char*)d_ws;

  // workspace layout (~6.5 MiB total)
  unsigned long long* keys = (unsigned long long*)ws;                 // B*M*8   = 4 MiB
  char* p = ws + (size_t)B_ * M_ * 8;
  float4* obA = (float4*)p;            p += (size_t)B_ * TOPK_ * 16;  // 1 MiB
  float4* bxA = (float4*)p;            p += (size_t)B_ * TOPK_ * 16;  // 1 MiB
  float*  scA = (float*)p;             p += (size_t)B_ * TOPK_ * 4;   // 256 KiB
  float*  clsA = (float*)p;                                           // 256 KiB

  // 1) scores -> sort keys (with padding)
  score_key_kernel<<<(B_ * M_) / 256, 256, 0, stream>>>(pred, keys);

  // 2) full descending bitonic sort per batch: 120 passes, L2-resident
  for (int k = 2; k <= M_; k <<= 1)
    for (int j = k >> 1; j >= 1; j >>= 1)
      bitonic_pass_kernel<<<(B_ * M_) / 256, 256, 0, stream>>>(keys, k, j);

  // 3) gather top-4096 candidates per batch
  gather_topk_kernel<<<(B_ * TOPK_) / 256, 256, 0, stream>>>(
      pred, keys, obA, bxA, scA, clsA);

  // 4) greedy NMS + compaction, one WGP-resident workgroup per image
  size_t smem = 65536 + 16384 + 4096 + 4096 + 260 * 4;  // ~90 KiB (<320 KiB/WGP)
  nms_kernel<<<B_, NMS_THREADS, smem, stream>>>(obA, bxA, scA, clsA, out);
}